// model2_fixed_emb_17016660427421
// MI455X (gfx1250) — compile-verified
//
#include <hip/hip_runtime.h>
#include <math.h>

// GCN stack for MI455X (gfx1250, wave32).
// Per layer:
//   [1] h = act(x) @ W   -> v_wmma_f32_16x16x4_f32; W staged in LDS (ds_load_b64 B operand),
//                           previous-layer ReLU fused into the A load (compile-time).
//   [2] out = b + h*dis^2     (self-loop + bias, non-atomic float4 stores)
//   [3] out[dst] += h[src]*norm over E real edges -> global_atomic_add_f32 (L2-resident)
//   [4] softmax(16) every 3rd layer, written out-of-place (or to d_out).
// deg/dis/norm recomputed every call (deterministic, no cross-call state).

typedef __attribute__((ext_vector_type(2))) float v2f;
typedef __attribute__((ext_vector_type(8))) float v8f;

#define WAVES_PER_BLOCK 4

// ---------------- degree / norm precompute ----------------

__global__ void init_deg_k(float* __restrict__ deg, int n) {
  int i = blockIdx.x * blockDim.x + threadIdx.x;
  if (i < n) deg[i] = 1.0f;  // self loop contributes 1
}

__global__ void count_deg_k(const int* __restrict__ dst, float* __restrict__ deg, int e) {
  int i = blockIdx.x * blockDim.x + threadIdx.x;
  if (i < e) unsafeAtomicAdd(&deg[dst[i]], 1.0f);
}

__global__ void to_dis_k(float* __restrict__ deg, int n) {
  int i = blockIdx.x * blockDim.x + threadIdx.x;
  if (i < n) deg[i] = rsqrtf(fmaxf(deg[i], 1.0f));
}

__global__ void norm_k(const int* __restrict__ src, const int* __restrict__ dst,
                       const float* __restrict__ dis, float* __restrict__ norm, int e) {
  int i = blockIdx.x * blockDim.x + threadIdx.x;
  if (i < e) norm[i] = dis[src[i]] * dis[dst[i]];
}

// ---------------- dense projection: h = act(x) @ W (fp32 WMMA) ----------------
// Compile-time shapes: loops fully unroll, ReLU is a single v_max_num (no select).
// One wave: 16-row stripe, loops 16-col tiles, K in steps of 4.
// A (16x4 f32): lanes 0-15 -> K={k,k+1} of row M=lane; lanes 16-31 -> K={k+2,k+3}.
// B staged in LDS pair-major: ldsW[(k>>1)*STRIDE + 2*col + (k&1)] -> one ds_load_b64.
// C/D (16x16 f32, 8 VGPRs): VGPR v -> row m0+v (lanes 0-15) / m0+8+v (lanes 16-31).

template <int DIN, int DOUT, bool RELU>
__global__ void gemm_wmma_f32(const float* __restrict__ x, const float* __restrict__ W,
                              float* __restrict__ h, int n_nodes) {
  constexpr int STRIDE = (DOUT << 1) + 2;  // +2 floats: no bank conflict between half-waves
  __shared__ float ldsW[(DIN / 2) * STRIDE];

  // cooperative W staging (all threads participate; no early return before barrier)
  for (int idx = threadIdx.x; idx < DIN * DOUT; idx += blockDim.x) {
    int k = idx / DOUT;
    int col = idx & (DOUT - 1);
    ldsW[(k >> 1) * STRIDE + (col << 1) + (k & 1)] = W[idx];
  }
  __syncthreads();

  int wave = threadIdx.x >> 5;
  int lane = threadIdx.x & 31;
  int m0 = (blockIdx.x * WAVES_PER_BLOCK + wave) * 16;
  if (m0 < n_nodes) {  // uniform per-wave; EXEC all-ones inside (WMMA requirement)
    int l15 = lane & 15;
    int khalf = lane >> 4;  // 0 -> K={k,k+1}, 1 -> K={k+2,k+3}
    int row = m0 + l15;

#pragma unroll
    for (int nt = 0; nt < DOUT; nt += 16) {
      v8f acc = {};
      int col = nt + l15;
#pragma unroll
      for (int k = 0; k < DIN; k += 4) {
        int ka = k + (khalf << 1);
        v2f a = *(const v2f*)(x + (size_t)row * DIN + ka);  // 8B aligned
        if (RELU) {
          a.x = fmaxf(a.x, 0.0f);
          a.y = fmaxf(a.y, 0.0f);
        }
        int kp = (k >> 1) + khalf;
        v2f b = *(const v2f*)(ldsW + kp * STRIDE + (col << 1));  // ds_load_b64
        acc = __builtin_amdgcn_wmma_f32_16x16x4_f32(
            /*neg_a=*/false, a, /*neg_b=*/false, b,
            /*c_mod=*/(short)0, acc, /*reuse_a=*/false, /*reuse_b=*/false);
      }
      int rbase = m0 + (khalf << 3);
#pragma unroll
      for (int v = 0; v < 8; ++v)
        h[(size_t)(rbase + v) * DOUT + col] = acc[v];
    }
  }
}

// ---------------- accumulator init: bias + self-loop (non-atomic) ----------------
// out[n][c] = b[c] + h[n][c] * dis[n]^2 ; one thread per (node, 4 channels)

__global__ void self_init_k(const float* __restrict__ h, const float* __restrict__ dis,
                            const float* __restrict__ b, float* __restrict__ out,
                            int n_nodes, int dout, int cshift) {
  size_t tid = (size_t)blockIdx.x * blockDim.x + threadIdx.x;
  int cg = (int)(tid & ((1u << cshift) - 1u));
  size_t nidx = tid >> cshift;
  if (nidx >= (size_t)n_nodes) return;
  float dv = dis[nidx];
  float nr = dv * dv;
  const float4 hv = *(const float4*)(h + nidx * dout + (cg << 2));
  const float4 bv = *(const float4*)(b + (cg << 2));
  float4 o;
  o.x = bv.x + hv.x * nr;
  o.y = bv.y + hv.y * nr;
  o.z = bv.z + hv.z * nr;
  o.w = bv.w + hv.w * nr;
  *(float4*)(out + nidx * dout + (cg << 2)) = o;
}

// ---------------- edge scatter: out[dst] += h[src] * norm ----------------
// One thread per (edge, 4 channels). Coalesced float4 gather, L2-resident fp32 atomics.

__global__ void scatter_k(const float* __restrict__ h, const float* __restrict__ norm,
                          const int* __restrict__ src, const int* __restrict__ dst,
                          float* __restrict__ out, int dout, int e, int cshift) {
  size_t tid = (size_t)blockIdx.x * blockDim.x + threadIdx.x;
  int cg = (int)(tid & ((1u << cshift) - 1u));
  size_t eidx = tid >> cshift;
  if (eidx >= (size_t)e) return;
  int s = src[eidx];
  int d = dst[eidx];
  float nr = norm[eidx];
  const float4 hv = *(const float4*)(h + (size_t)s * dout + (cg << 2));
  float* o = out + (size_t)d * dout + (cg << 2);
  unsafeAtomicAdd(o + 0, hv.x * nr);
  unsafeAtomicAdd(o + 1, hv.y * nr);
  unsafeAtomicAdd(o + 2, hv.z * nr);
  unsafeAtomicAdd(o + 3, hv.w * nr);
}

// ---------------- softmax over 16 channels ----------------

__global__ void softmax16_k(const float* __restrict__ in, float* __restrict__ out, int n) {
  int i = blockIdx.x * blockDim.x + threadIdx.x;
  if (i >= n) return;
  float v[16];
  const float* p = in + (size_t)i * 16;
#pragma unroll
  for (int j = 0; j < 16; ++j) v[j] = p[j];
  float m = v[0];
#pragma unroll
  for (int j = 1; j < 16; ++j) m = fmaxf(m, v[j]);
  float s = 0.0f;
#pragma unroll
  for (int j = 0; j < 16; ++j) { v[j] = __expf(v[j] - m); s += v[j]; }
  float inv = 1.0f / s;
  float* o = out + (size_t)i * 16;
#pragma unroll
  for (int j = 0; j < 16; ++j) o[j] = v[j] * inv;
}

// ---------------- host orchestration ----------------

extern "C" void kernel_launch(void* const* d_in, const int* in_sizes, int n_in,
                              void* d_out, int out_size, void* d_ws, size_t ws_size,
                              hipStream_t stream) {
  const float* x  = (const float*)d_in[0];
  const int*   ei = (const int*)d_in[1];
  const float* W1 = (const float*)d_in[2];
  const float* b1 = (const float*)d_in[3];
  const float* W2 = (const float*)d_in[4];
  const float* b2 = (const float*)d_in[5];
  const float* W3 = (const float*)d_in[6];
  const float* b3 = (const float*)d_in[7];

  const int F = 16;
  int n_nodes = in_sizes[0] / F;  // 100000 (divisible by 16)
  int E = in_sizes[1] / 2;        // 3200000
  const int* srcI = ei;
  const int* dstI = ei + E;

  // workspace carve-up (~90 MB): 3 ping-pong N*64 buffers, deg/dis, norm(E)
  float* bufA = (float*)d_ws;
  float* bufB = bufA + (size_t)n_nodes * 64;
  float* bufC = bufB + (size_t)n_nodes * 64;
  float* dis  = bufC + (size_t)n_nodes * 64;
  float* norm = dis + n_nodes;

  const int T = 256;
  // --- degree / norm precompute (every call; deterministic) ---
  init_deg_k<<<(n_nodes + T - 1) / T, T, 0, stream>>>(dis, n_nodes);
  count_deg_k<<<(E + T - 1) / T, T, 0, stream>>>(dstI, dis, E);
  to_dis_k<<<(n_nodes + T - 1) / T, T, 0, stream>>>(dis, n_nodes);
  norm_k<<<(E + T - 1) / T, T, 0, stream>>>(srcI, dstI, dis, norm, E);

  const float* xin = x;  // current layer input (pre-activation; gemm applies ReLU)
  float* xbuf = bufA;    // free buffer
  float* hbuf = bufB;    // projection result
  float* obuf = bufC;    // aggregation accumulator

  int tiles = (n_nodes + 15) / 16;
  int gemm_blocks = (tiles + WAVES_PER_BLOCK - 1) / WAVES_PER_BLOCK;
  const int GT = WAVES_PER_BLOCK * 32;

  for (int blk = 0; blk < 5; ++blk) {
    const float* Wb1 = W1 + (size_t)blk * 16 * 64;
    const float* Wb2 = W2 + (size_t)blk * 64 * 32;
    const float* Wb3 = W3 + (size_t)blk * 32 * 16;
    const float* bs[3] = { b1 + (size_t)blk * 64,
                           b2 + (size_t)blk * 32,
                           b3 + (size_t)blk * 16 };
    const int douts[3]  = {64, 32, 16};
    const int cshifts[3] = {4, 3, 2};  // log2(dout/4)

    for (int l = 0; l < 3; ++l) {
      int dout = douts[l];
      int cshift = cshifts[l];

      // [1] projection (WMMA fp32, LDS-staged W, compile-time fused ReLU)
      if (l == 0)
        gemm_wmma_f32<16, 64, false><<<gemm_blocks, GT, 0, stream>>>(xin, Wb1, hbuf, n_nodes);
      else if (l == 1)
        gemm_wmma_f32<64, 32, true><<<gemm_blocks, GT, 0, stream>>>(xin, Wb2, hbuf, n_nodes);
      else
        gemm_wmma_f32<32, 16, true><<<gemm_blocks, GT, 0, stream>>>(xin, Wb3, hbuf, n_nodes);

      // [2] accumulator = bias + self-loop contribution (non-atomic)
      size_t sithreads = (size_t)n_nodes << cshift;
      self_init_k<<<(unsigned)((sithreads + T - 1) / T), T, 0, stream>>>(
          hbuf, dis, bs[l], obuf, n_nodes, dout, cshift);

      // [3] edge scatter over real edges only
      size_t sthreads = (size_t)E << cshift;
      scatter_k<<<(unsigned)((sthreads + T - 1) / T), T, 0, stream>>>(
          hbuf, norm, srcI, dstI, obuf, dout, E, cshift);

      // [4] activation / buffer rotation
      if (l < 2) {
        // ReLU deferred into next gemm's A load
        xin = obuf;
        float* t = xbuf; xbuf = obuf; obuf = t;
      } else {
        float* dstp = (blk == 4) ? (float*)d_out : xbuf;  // softmax out-of-place
        softmax16_k<<<(n_nodes + T - 1) / T, T, 0, stream>>>(obuf, dstp, n_nodes);
        xin = dstp;
        // roles unchanged: xbuf holds data (xin), obuf free for next layer
      }
    }
  }
}